// Aggregator_8040178778538
// MI455X (gfx1250) — compile-verified
//
#include <hip/hip_runtime.h>
#include <hip/hip_bf16.h>
#include <stdint.h>

#define CHANNEL 64
#define N_REL   24

typedef __attribute__((ext_vector_type(4))) unsigned int v4u;
typedef __attribute__((ext_vector_type(4))) int          v4i;
typedef __attribute__((ext_vector_type(8))) int          v8i;

__device__ __forceinline__ void atomic_add_f32(float* p, float v) {
    // relaxed, device scope -> should lower to global_atomic_add_f32 (no-return)
    __hip_atomic_fetch_add(p, v, __ATOMIC_RELAXED, __HIP_MEMORY_SCOPE_AGENT);
}

__global__ __launch_bounds__(256) void rgcn_edge_agg_kernel(
    const float* __restrict__ all_emb,       // [N_NODES, 64]
    const long long* __restrict__ edge_head, // [E]
    const long long* __restrict__ edge_tail, // [E]
    const long long* __restrict__ edge_type, // [E]
    const float* __restrict__ weight,        // [24, 64]
    const float* __restrict__ aug_w,         // [E]
    float* __restrict__ out,                 // [N_NODES, 64] (pre-zeroed)
    int n_edges)
{
    __shared__ __align__(16) float lds_w[N_REL * CHANNEL]; // 6 KB relation table

    // ---- Stage weight table into LDS via the Tensor Data Mover (CDNA5) ----
#if __has_builtin(__builtin_amdgcn_tensor_load_to_lds)
    if (threadIdx.x < 32) { // one wave issues the descriptor (EXEC ignored by TDM)
        uint64_t ga = (uint64_t)weight;
        uint32_t lofs = (uint32_t)(uintptr_t)lds_w;   // low 32 bits = LDS byte offset
        v4u g0;
        g0[0] = 1u;                                   // count=1, user descriptor
        g0[1] = lofs;                                 // lds_addr
        g0[2] = (uint32_t)ga;                         // global_addr[31:0]
        g0[3] = (uint32_t)((ga >> 32) & 0x01FFFFFFu)  // global_addr[56:32]
              | 0x80000000u;                          // type = 2 ("image")
        v8i g1;
        g1[0] = 0x00020000;                           // data_size = 4 bytes
        g1[1] = (int)(1536u << 16);                   // tensor_dim0 = 1536 (lo16)
        g1[2] = (int)(1u << 16);                      // dim0 hi=0 | tensor_dim1 = 1
        g1[3] = (int)(1536u << 16);                   // tile_dim0 = 1536 (1-D tile)
        g1[4] = 0;                                    // tile_dim1/2 unused
        g1[5] = 1536;                                 // tensor_dim0_stride
        g1[6] = 0;
        g1[7] = 0;
        v4i gz = {0, 0, 0, 0};
#if __clang_major__ >= 23
        v8i gz8 = {0, 0, 0, 0, 0, 0, 0, 0};
        __builtin_amdgcn_tensor_load_to_lds(g0, g1, gz, gz, gz8, 0);
#else
        __builtin_amdgcn_tensor_load_to_lds(g0, g1, gz, gz, 0);
#endif
        __builtin_amdgcn_s_wait_tensorcnt((short)0);
    }
#else
    for (int i = threadIdx.x; i < N_REL * CHANNEL; i += blockDim.x)
        lds_w[i] = weight[i];
#endif
    __syncthreads();

    const int lane  = threadIdx.x & 31;
    const int wid   = (blockIdx.x * blockDim.x + threadIdx.x) >> 5;
    const int nwave = (gridDim.x * blockDim.x) >> 5;
    const int c0    = lane * 2;                       // 2 channels per lane

    // Each wave consumes edges in batches of 32: lane L preloads edge base+L's
    // metadata (coalesced), then the wave iterates the batch via shuffles.
    for (long long base = (long long)wid * 32; base < n_edges;
         base += (long long)nwave * 32) {
        long long e = base + lane;
        int h = 0, t = 0, r = 0;
        float aw = 0.0f;
        if (e < n_edges) {
            h  = (int)edge_head[e];
            t  = (int)edge_tail[e];
            r  = (int)edge_type[e];
            aw = aug_w[e];
        }
        long long rem = (long long)n_edges - base;
        int cnt = rem < 32 ? (int)rem : 32;

        for (int j = 0; j < cnt; ++j) {
            int   hj  = __shfl(h,  j, 32);
            int   tj  = __shfl(t,  j, 32);
            int   rj  = __shfl(r,  j, 32);
            float awj = __shfl(aw, j, 32);

            const float2 a = *(const float2*)(all_emb + (size_t)tj * CHANNEL + c0);
            const float2 w = *(const float2*)(lds_w + rj * CHANNEL + c0);

            float2 v;
            v.x = a.x * w.x * awj;
            v.y = a.y * w.y * awj;

            float* o = out + (size_t)hj * CHANNEL + c0;
            atomic_add_f32(o,     v.x);
            atomic_add_f32(o + 1, v.y);
        }
    }
}

extern "C" void kernel_launch(void* const* d_in, const int* in_sizes, int n_in,
                              void* d_out, int out_size, void* d_ws, size_t ws_size,
                              hipStream_t stream) {
    const float*     all_emb = (const float*)d_in[0];
    const long long* eidx    = (const long long*)d_in[1]; // [2, E] row-major
    const long long* etype   = (const long long*)d_in[2];
    const float*     weight  = (const float*)d_in[3];
    const float*     aug_w   = (const float*)d_in[4];
    float*           out     = (float*)d_out;

    const int nE = in_sizes[2]; // edge_type element count == number of edges

    // Output is accumulated with atomics: zero it every call (capturable).
    hipMemsetAsync(d_out, 0, (size_t)out_size * sizeof(float), stream);

    dim3 block(256);
    dim3 grid(2048); // 16384 wave32s: saturates WGPs, amortizes the 6KB TDM load
    rgcn_edge_agg_kernel<<<grid, block, 0, stream>>>(
        all_emb, eidx, eidx + nE, etype, weight, aug_w, out, nE);
}